// TrajectoryOptimizedMRI_31439160607185
// MI455X (gfx1250) — compile-verified
//
#include <hip/hip_runtime.h>
#include <math.h>

// ---------------------------------------------------------------------------
// TrajectoryOptimizedMRI on MI455X (gfx1250, wave32, WMMA).
// Fourier sampling + 3-layer MLP, all matrix math via V_WMMA_F32_16X16X4_F32
// (fp32-exact, matches the fp32/complex64 reference).
// Roofline: W3 stream (268 MB) dominates -> ~11.5 us at 23.3 TB/s; total
// ~6.5 GFLOP is far below WMMA ceiling, so keep full fp32 precision.
// ---------------------------------------------------------------------------

typedef __attribute__((ext_vector_type(2))) float v2f;
typedef __attribute__((ext_vector_type(8))) float v8f;

#define T_PTS 1024
#define H_DIM 256
#define W_DIM 256
#define D1_N  512
#define D2_N  1024
#define B_SZ  16
#define TWO_PI 6.2831853071795864769f

// D = A(16x4) * B(4x16) + C, fp32.
__device__ __forceinline__ v8f wmma_f32(v2f a, v2f b, v8f c) {
    return __builtin_amdgcn_wmma_f32_16x16x4_f32(
        /*neg_a=*/false, a, /*neg_b=*/false, b,
        /*c_mod=*/(short)0, c, /*reuse_a=*/false, /*reuse_b=*/false);
}

// ---------------------------------------------------------------------------
// Kernel 1: phase tables  px[t,w] = exp(-i*2pi*kx[t]*xx[w]),
//                         py[t,h] = exp(-i*2pi*ky[t]*yy[h])
// plus copy of k_points into the output tail.
// ---------------------------------------------------------------------------
__global__ __launch_bounds__(256)
void build_tables(const float* __restrict__ kp,
                  float* __restrict__ pxr, float* __restrict__ pxi,
                  float* __restrict__ pyr, float* __restrict__ pyi,
                  float* __restrict__ kout) {
    int idx = blockIdx.x * 256 + threadIdx.x;
    if (idx < 2 * T_PTS) kout[idx] = kp[idx];   // k_points passthrough output

    const int NPX = T_PTS * W_DIM;
    if (idx < NPX) {
        int t = idx >> 8, j = idx & 255;
        float xx = -0.5f + (float)j * (1.0f / (W_DIM - 1));
        float th = TWO_PI * kp[2 * t + 0] * xx;
        float s, c; sincosf(th, &s, &c);
        pxr[idx] = c; pxi[idx] = -s;
    } else {
        int i2 = idx - NPX;
        if (i2 < T_PTS * H_DIM) {
            int t = i2 >> 8, j = i2 & 255;
            float yy = -0.5f + (float)j * (1.0f / (H_DIM - 1));
            float th = TWO_PI * kp[2 * t + 1] * yy;
            float s, c; sincosf(th, &s, &c);
            pyr[i2] = c; pyi[i2] = -s;
        }
    }
}

// ---------------------------------------------------------------------------
// Kernel 2: fused Fourier sampling.
// Block = (b, t-tile of 16).  16 waves; wave w owns w-columns [16w,16w+16).
//   c[t,w] = sum_h py[t,h] * x[b,h,w]        (two fp32 WMMA GEMMs, K=H=256)
//   y[b,t] = sum_w c[t,w] * px[t,w]          (complex, shfl+LDS reduce)
// Writes y (complex interleaved) and feat = [Re(y) | Im(y)].
// ---------------------------------------------------------------------------
__global__ __launch_bounds__(512)
void fourier_kernel(const float* __restrict__ x,
                    const float* __restrict__ pxr, const float* __restrict__ pxi,
                    const float* __restrict__ pyr, const float* __restrict__ pyi,
                    float* __restrict__ outy, float* __restrict__ feat) {
    __shared__ float yr[16];
    __shared__ float yi[16];

    const int b   = blockIdx.x >> 6;        // /(T/16)
    const int t0  = (blockIdx.x & 63) * 16;
    const int wav = threadIdx.x >> 5;
    const int lane = threadIdx.x & 31;
    const int ln15 = lane & 15;             // = M row of A frag = N col of B/C frag
    const int hi   = lane >> 4;

    if (threadIdx.x < 16) { yr[threadIdx.x] = 0.0f; yi[threadIdx.x] = 0.0f; }
    __syncthreads();

    const int w0 = wav * 16;
    const float* Ar = pyr + t0 * H_DIM;            // (16 x H) row-major
    const float* Ai = pyi + t0 * H_DIM;
    const float* Bx = x + (size_t)b * H_DIM * W_DIM; // (H x W) row-major

    v8f cr = {}; v8f ci = {};
    for (int k0 = 0; k0 < H_DIM; k0 += 4) {
        const int kb = k0 + 2 * hi;
        v2f ar = *(const v2f*)(Ar + ln15 * H_DIM + kb);   // A[m][kb], A[m][kb+1]
        v2f ai = *(const v2f*)(Ai + ln15 * H_DIM + kb);
        v2f bb;
        bb.x = Bx[(size_t)kb * W_DIM + w0 + ln15];        // B[kb][n]
        bb.y = Bx[(size_t)(kb + 1) * W_DIM + w0 + ln15];  // B[kb+1][n]
        cr = wmma_f32(ar, bb, cr);
        ci = wmma_f32(ai, bb, ci);
    }

    // Multiply by px and reduce over w.  C frag: VGPR j holds row m = j + 8*hi,
    // col n = ln15.
    for (int j = 0; j < 8; ++j) {
        const int m  = j + 8 * hi;
        const int tg = t0 + m;
        const float pr = pxr[tg * W_DIM + w0 + ln15];
        const float pi = pxi[tg * W_DIM + w0 + ln15];
        float vr = cr[j] * pr - ci[j] * pi;
        float vi = cr[j] * pi + ci[j] * pr;
        #pragma unroll
        for (int off = 8; off > 0; off >>= 1) {           // reduce the 16-lane half
            vr += __shfl_xor(vr, off, 32);
            vi += __shfl_xor(vi, off, 32);
        }
        if (ln15 == 0) {
            atomicAdd(&yr[m], vr);
            atomicAdd(&yi[m], vi);
        }
    }
    __syncthreads();

    if (threadIdx.x < 16) {
        const int t = t0 + threadIdx.x;
        const float r = yr[threadIdx.x], im = yi[threadIdx.x];
        outy[((size_t)b * T_PTS + t) * 2 + 0] = r;        // complex64 interleave
        outy[((size_t)b * T_PTS + t) * 2 + 1] = im;
        feat[(size_t)b * 2 * T_PTS + t]         = r;
        feat[(size_t)b * 2 * T_PTS + T_PTS + t] = im;
    }
}

// ---------------------------------------------------------------------------
// Kernels 3-5: C(16 x N) = relu?(A(16 x K) @ W(K x N) + bias)
// One 16-column tile per wave; K-loop of fp32 WMMA.  Layer 3 streams W3
// exactly once (memory-bound, 268 MB).
// ---------------------------------------------------------------------------
template <bool RELU>
__global__ __launch_bounds__(256)
void mlp_gemm(const float* __restrict__ A, const float* __restrict__ Wt,
              const float* __restrict__ bias, float* __restrict__ C,
              int K, int N) {
    const int wav = threadIdx.x >> 5;
    const int lane = threadIdx.x & 31;
    const int ln15 = lane & 15;
    const int hi   = lane >> 4;
    const int n0 = (blockIdx.x * 8 + wav) * 16;
    if (n0 >= N) return;

    v8f acc = {};
    for (int k0 = 0; k0 < K; k0 += 4) {
        const int kb = k0 + 2 * hi;
        v2f a = *(const v2f*)(A + ln15 * K + kb);
        v2f bb;
        bb.x = Wt[(size_t)kb * N + n0 + ln15];
        bb.y = Wt[(size_t)(kb + 1) * N + n0 + ln15];
        acc = wmma_f32(a, bb, acc);
    }

    const float bv = bias[n0 + ln15];
    #pragma unroll
    for (int j = 0; j < 8; ++j) {
        const int m = j + 8 * hi;
        float v = acc[j] + bv;
        if (RELU) v = fmaxf(v, 0.0f);
        C[(size_t)m * N + n0 + ln15] = v;
    }
}

// ---------------------------------------------------------------------------
// Launch
// ---------------------------------------------------------------------------
extern "C" void kernel_launch(void* const* d_in, const int* in_sizes, int n_in,
                              void* d_out, int out_size, void* d_ws, size_t ws_size,
                              hipStream_t stream) {
    const float* x  = (const float*)d_in[0];
    const float* kp = (const float*)d_in[1];
    const float* W1 = (const float*)d_in[2];
    const float* b1 = (const float*)d_in[3];
    const float* W2 = (const float*)d_in[4];
    const float* b2 = (const float*)d_in[5];
    const float* W3 = (const float*)d_in[6];
    const float* b3 = (const float*)d_in[7];
    float* out = (float*)d_out;

    // Workspace layout (floats), ~4.3 MB total
    float* ws   = (float*)d_ws;
    float* pxr  = ws;                         // T*W
    float* pxi  = pxr  + T_PTS * W_DIM;       // T*W
    float* pyr  = pxi  + T_PTS * W_DIM;       // T*H
    float* pyi  = pyr  + T_PTS * H_DIM;       // T*H
    float* feat = pyi  + T_PTS * H_DIM;       // B * 2T
    float* h1   = feat + B_SZ * 2 * T_PTS;    // B * D1
    float* h2   = h1   + B_SZ * D1_N;         // B * D2

    // Output layout: x_hat (B*H*W) | y (B*T complex interleaved) | k_points
    float* xhat = out;
    float* yout = out + (size_t)B_SZ * H_DIM * W_DIM;
    float* kout = yout + (size_t)B_SZ * T_PTS * 2;

    // K1: phase tables (T*(W+H) = 524288 points) + k_points copy
    build_tables<<<(T_PTS * (W_DIM + H_DIM)) / 256, 256, 0, stream>>>(
        kp, pxr, pxi, pyr, pyi, kout);

    // K2: fused Fourier sampling -> y, feat
    fourier_kernel<<<B_SZ * (T_PTS / 16), 512, 0, stream>>>(
        x, pxr, pxi, pyr, pyi, yout, feat);

    // K3: feat(16x2048) @ W1 -> h1 (relu)
    mlp_gemm<true><<<D1_N / (16 * 8), 256, 0, stream>>>(feat, W1, b1, h1, 2 * T_PTS, D1_N);
    // K4: h1(16x512) @ W2 -> h2 (relu)
    mlp_gemm<true><<<D2_N / (16 * 8), 256, 0, stream>>>(h1, W2, b2, h2, D1_N, D2_N);
    // K5: h2(16x1024) @ W3 -> x_hat (streams 268 MB of W3; HBM-bound)
    mlp_gemm<false><<<(H_DIM * W_DIM) / (16 * 8), 256, 0, stream>>>(
        h2, W3, b3, xhat, D2_N, H_DIM * W_DIM);
}